// Edgeconvf_687194767628
// MI455X (gfx1250) — compile-verified
//
#include <hip/hip_runtime.h>

// CDNA5 / gfx1250, wave32. FP32 WMMA (16x16x4) keeps full fp32 precision
// vs the reference while using the matrix pipe.

typedef float v2f __attribute__((ext_vector_type(2)));
typedef float v8f __attribute__((ext_vector_type(8)));

#define XS_STRIDE 68   // 64 + 4 pad  (stride % 16 == 4 -> conflict-free b64 A reads)
#define FS_STRIDE 100  // 96 + 4 pad  (same property)

// ---------------------------------------------------------------------------
// Kernel 1: proj[N,64] = x[N,64] @ w1^T + b1      (w1 is [64 out][64 in])
// One block = 128 threads = 4 waves; each wave -> 16 rows x 64 cols.
// ---------------------------------------------------------------------------
__global__ void __launch_bounds__(128)
proj_kernel(const float* __restrict__ x, const float* __restrict__ w1,
            const float* __restrict__ b1, float* __restrict__ proj, int N)
{
    extern __shared__ float smem[];
    float* xs = smem;                    // 64 * XS_STRIDE floats (A tile)
    float* bf = smem + 64 * XS_STRIDE;   // 16 kt * 4 nt * 64    (B fragments)

    const int tid  = threadIdx.x;
    const int lane = tid & 31;
    const int wv   = tid >> 5;
    const int r0   = blockIdx.x * 64;

    // Stage w1 as WMMA B fragments: B[k][n] = w1[n*64 + k].
    // Fragment layout (mirrors documented 32-bit A 16x4 layout):
    //   VGPR0: lanes0-15 -> K=kt*4+0, lanes16-31 -> K=kt*4+2
    //   VGPR1: lanes0-15 -> K=kt*4+1, lanes16-31 -> K=kt*4+3
    #pragma unroll
    for (int i = 0; i < 32; ++i) {
        int idx = tid + i * 128;             // 0..4095
        int kt = idx >> 8;
        int nt = (idx >> 6) & 3;
        int ln = (idx >> 1) & 31;
        int j  = idx & 1;
        int k  = kt * 4 + ((ln >= 16) ? 2 : 0) + j;
        int n  = nt * 16 + (ln & 15);
        bf[idx] = w1[n * 64 + k];
    }
    // Stage 64 rows of x, fully coalesced (float2 per thread per iter)
    #pragma unroll
    for (int i = 0; i < 16; ++i) {
        int idx2 = tid + i * 128;            // 0..2047 float2 slots
        int r  = idx2 >> 5;                  // 32 float2 per row
        int c2 = idx2 & 31;
        int g  = r0 + r;
        v2f v = {0.f, 0.f};
        if (g < N) v = *(const v2f*)(x + (size_t)g * 64 + c2 * 2);
        *(v2f*)(xs + r * XS_STRIDE + c2 * 2) = v;
    }
    __syncthreads();

    const int m0   = wv * 16;
    const int nlo  = lane & 15;
    const int koff = (lane >= 16) ? 2 : 0;
    const float* abase = xs + (m0 + nlo) * XS_STRIDE + koff;

    v8f acc[4];
    #pragma unroll
    for (int nt = 0; nt < 4; ++nt) {
        float bias = b1[nt * 16 + nlo];      // lane selects N column, same for all 8 M rows
        #pragma unroll
        for (int c = 0; c < 8; ++c) acc[nt][c] = bias;
    }

    #pragma unroll
    for (int kt = 0; kt < 16; ++kt) {
        v2f a = *(const v2f*)(abase + kt * 4);
        #pragma unroll
        for (int nt = 0; nt < 4; ++nt) {
            v2f b = *(const v2f*)(bf + ((kt * 4 + nt) * 32 + lane) * 2);
            acc[nt] = __builtin_amdgcn_wmma_f32_16x16x4_f32(
                false, a, false, b, (short)0, acc[nt], false, false);
        }
    }

    const int mhi = (lane >= 16) ? 8 : 0;
    #pragma unroll
    for (int nt = 0; nt < 4; ++nt) {
        int n = nt * 16 + nlo;
        #pragma unroll
        for (int v = 0; v < 8; ++v) {
            int m = r0 + m0 + v + mhi;       // C layout: VGPR v -> M row
            if (m < N) proj[(size_t)m * 64 + n] = acc[nt][v];
        }
    }
}

// ---------------------------------------------------------------------------
// Kernel 2: per-edge  out[e] = [relu(proj[src]+proj[dst]) | ea | ef] @ w2^T + b2
// proj (25.6 MB) is L2-resident -> random gathers hit L2, not HBM.
// One block = 128 threads = 4 waves; each wave -> 16 edges x 64 out channels.
// ---------------------------------------------------------------------------
__global__ void __launch_bounds__(128)
edge_kernel(const float* __restrict__ proj, const int* __restrict__ edge_index,
            const float* __restrict__ edge_attr, const float* __restrict__ edge_f,
            const float* __restrict__ w2, const float* __restrict__ b2,
            float* __restrict__ out, int E)
{
    extern __shared__ float smem[];
    float* fs = smem;                    // 64 * FS_STRIDE floats (feature tile)
    float* bf = smem + 64 * FS_STRIDE;   // 24 kt * 4 nt * 64    (B fragments)

    const int tid  = threadIdx.x;
    const int lane = tid & 31;
    const int wv   = tid >> 5;

    // Stage w2 as WMMA B fragments: B[k][n] = w2[n*96 + k]
    #pragma unroll
    for (int i = 0; i < 48; ++i) {
        int idx = tid + i * 128;             // 0..6143
        int kt = idx >> 8;
        int nt = (idx >> 6) & 3;
        int ln = (idx >> 1) & 31;
        int j  = idx & 1;
        int k  = kt * 4 + ((ln >= 16) ? 2 : 0) + j;
        int n  = nt * 16 + (ln & 15);
        bf[idx] = w2[n * 96 + k];
    }

    const int e0  = blockIdx.x * 64 + wv * 16;
    const int nlo = lane & 15;
    const int lr0 = wv * 16;
    const int* srcp = edge_index;        // edge_index[0] row
    const int* dstp = edge_index + E;    // edge_index[1] row

    // lanes 0-15 hold this wave's 16 src indices, lanes 16-31 the dst indices
    int eL = e0 + nlo;
    const int* ip = (lane < 16) ? (srcp + eL) : (dstp + eL);
    int idxval = (eL < E) ? *ip : 0;

    // Build 16 feature rows in LDS: cols 0-63 = relu(proj_s + proj_d),
    // cols 64-79 = edge_attr, cols 80-95 = edge_f.
    for (int i = 0; i < 16; ++i) {
        int e = e0 + i;
        bool act = (e < E);
        int s = __shfl(idxval, i, 32);
        int d = __shfl(idxval, i + 16, 32);
        v2f h = {0.f, 0.f};
        if (act) {
            // one fully coalesced 256B row read per load (32 lanes x 8B)
            v2f ps = *(const v2f*)(proj + (size_t)s * 64 + lane * 2);
            v2f pd = *(const v2f*)(proj + (size_t)d * 64 + lane * 2);
            h = ps + pd;
            h.x = fmaxf(h.x, 0.f);
            h.y = fmaxf(h.y, 0.f);
        }
        *(v2f*)(fs + (lr0 + i) * FS_STRIDE + lane * 2) = h;
        // cols 64+lane: lanes 0-15 <- edge_attr[e][lane], 16-31 <- edge_f[e][lane-16]
        float ev = 0.f;
        if (act) {
            const float* ep = (lane < 16) ? (edge_attr + (size_t)e * 16 + lane)
                                          : (edge_f    + (size_t)e * 16 + (lane - 16));
            ev = *ep;
        }
        fs[(lr0 + i) * FS_STRIDE + 64 + lane] = ev;
    }
    __syncthreads();

    const int koff = (lane >= 16) ? 2 : 0;
    const float* abase = fs + (lr0 + nlo) * FS_STRIDE + koff;

    v8f acc[4];
    #pragma unroll
    for (int nt = 0; nt < 4; ++nt) {
        float bias = b2[nt * 16 + nlo];
        #pragma unroll
        for (int c = 0; c < 8; ++c) acc[nt][c] = bias;
    }

    #pragma unroll
    for (int kt = 0; kt < 24; ++kt) {       // K = 96 = 24 * 4
        v2f a = *(const v2f*)(abase + kt * 4);
        #pragma unroll
        for (int nt = 0; nt < 4; ++nt) {
            v2f b = *(const v2f*)(bf + ((kt * 4 + nt) * 32 + lane) * 2);
            acc[nt] = __builtin_amdgcn_wmma_f32_16x16x4_f32(
                false, a, false, b, (short)0, acc[nt], false, false);
        }
    }

    const int mhi = (lane >= 16) ? 8 : 0;
    #pragma unroll
    for (int nt = 0; nt < 4; ++nt) {
        int n = nt * 16 + nlo;
        #pragma unroll
        for (int v = 0; v < 8; ++v) {
            int m = e0 + v + mhi;
            if (m < E) out[(size_t)m * 64 + n] = acc[nt][v];
        }
    }
}

// ---------------------------------------------------------------------------
extern "C" void kernel_launch(void* const* d_in, const int* in_sizes, int n_in,
                              void* d_out, int out_size, void* d_ws, size_t ws_size,
                              hipStream_t stream)
{
    const float* x  = (const float*)d_in[0];
    const int*   ei = (const int*)  d_in[1];   // [2, E] int32 (src row, then dst row)
    const float* ea = (const float*)d_in[2];   // [E, 16]
    const float* ef = (const float*)d_in[3];   // [E, 16]
    const float* w1 = (const float*)d_in[4];   // [64, 64]
    const float* b1 = (const float*)d_in[5];   // [64]
    const float* w2 = (const float*)d_in[6];   // [64, 96]
    const float* b2 = (const float*)d_in[7];   // [64]
    float* out = (float*)d_out;

    const int N = in_sizes[0] / 64;            // nodes
    const int E = in_sizes[2] / 16;            // edges

    float* proj = (float*)d_ws;                // N*64 floats (~25.6 MB, L2-resident)

    const size_t sm1 = (size_t)(64 * XS_STRIDE + 16 * 4 * 64) * sizeof(float);
    const size_t sm2 = (size_t)(64 * FS_STRIDE + 24 * 4 * 64) * sizeof(float);
    const int g1 = (N + 63) / 64;
    const int g2 = (E + 63) / 64;

    proj_kernel<<<g1, 128, sm1, stream>>>(x, w1, b1, proj, N);
    edge_kernel<<<g2, 128, sm2, stream>>>(proj, ei, ea, ef, w2, b2, out, E);
}